// Skeleton_3307124818535
// MI455X (gfx1250) — compile-verified
//
#include <hip/hip_runtime.h>

#define NJ      24
#define TPB     256
#define ELEM_F  144      // floats per element: 24 joints * 6
#define STRIDE  148      // padded floats per element in LDS (bank-conflict mitigation)
#define OUT_F   72       // floats per element out: 24 * 3
#define CHUNKS  36       // 16-byte chunks per thread to stage (ELEM_F/4 per element)

static constexpr int kPar[NJ] = {-1, 0, 0, 0, 1, 2, 3, 4, 5, 6, 7, 8,
                                  9, 9, 9, 12, 13, 14, 16, 17, 18, 19, 20, 21};

// ---- CDNA5 async global->LDS copy (ASYNCcnt-tracked) -----------------------
typedef int v4i __attribute__((vector_size(16)));
typedef __attribute__((address_space(1))) v4i* gptr_v4i;
typedef __attribute__((address_space(3))) v4i* lptr_v4i;

__device__ __forceinline__ void async_copy_b128(float* dstLds, const float* srcGlb) {
#if defined(__has_builtin) && __has_builtin(__builtin_amdgcn_global_load_async_to_lds_b128)
  __builtin_amdgcn_global_load_async_to_lds_b128(
      (gptr_v4i)(void*)srcGlb,   // global source (AS1)
      (lptr_v4i)(void*)dstLds,   // LDS destination (AS3)
      0, 0);
#else
  unsigned l = (unsigned)(unsigned long long)dstLds;   // low 32 bits = LDS offset
  asm volatile("global_load_async_to_lds_b128 %0, %1, off"
               :: "v"(l), "v"(srcGlb) : "memory");
#endif
}

__device__ __forceinline__ void async_wait0() {
#if defined(__has_builtin) && __has_builtin(__builtin_amdgcn_s_wait_asynccnt)
  __builtin_amdgcn_s_wait_asynccnt(0);
#else
  asm volatile("s_wait_asynccnt 0" ::: "memory");
#endif
}

// ---- Forward kinematics for one element ------------------------------------
__device__ __forceinline__ void fk_elem(const float* __restrict__ A,
                                        const float* __restrict__ X,
                                        float* __restrict__ op) {
  float R[NJ][9];
  float T[NJ][3];
#pragma unroll
  for (int c = 0; c < NJ; ++c) {
    const float a1x = A[c * 6 + 0], a1y = A[c * 6 + 1], a1z = A[c * 6 + 2];
    const float a2x = A[c * 6 + 3], a2y = A[c * 6 + 4], a2z = A[c * 6 + 5];

    const float s1 = __builtin_amdgcn_rsqf(a1x * a1x + a1y * a1y + a1z * a1z);
    const float r00 = a1x * s1, r01 = a1y * s1, r02 = a1z * s1;          // b1
    const float d = r00 * a2x + r01 * a2y + r02 * a2z;
    float b2x = a2x - d * r00, b2y = a2y - d * r01, b2z = a2z - d * r02;
    const float s2 = __builtin_amdgcn_rsqf(b2x * b2x + b2y * b2y + b2z * b2z);
    const float r10 = b2x * s2, r11 = b2y * s2, r12 = b2z * s2;          // b2
    const float r20 = r01 * r12 - r02 * r11;                             // b3 = b1 x b2
    const float r21 = r02 * r10 - r00 * r12;
    const float r22 = r00 * r11 - r01 * r10;

    const int p = kPar[c];
    float tx, ty, tz;
    if (c == 0) {
      tx = X[0]; ty = X[1]; tz = X[2];
    } else {
      tx = X[c * 3 + 0] - X[p * 3 + 0];
      ty = X[c * 3 + 1] - X[p * 3 + 1];
      tz = X[c * 3 + 2] - X[p * 3 + 2];
    }
    const float ltx = r00 * tx + r01 * ty + r02 * tz;   // r @ t_local
    const float lty = r10 * tx + r11 * ty + r12 * tz;
    const float ltz = r20 * tx + r21 * ty + r22 * tz;

    if (c == 0) {
      R[0][0] = r00; R[0][1] = r01; R[0][2] = r02;
      R[0][3] = r10; R[0][4] = r11; R[0][5] = r12;
      R[0][6] = r20; R[0][7] = r21; R[0][8] = r22;
      T[0][0] = ltx; T[0][1] = lty; T[0][2] = ltz;
    } else {
#pragma unroll
      for (int i = 0; i < 3; ++i) {
        const float pi0 = R[p][i * 3 + 0];
        const float pi1 = R[p][i * 3 + 1];
        const float pi2 = R[p][i * 3 + 2];
        R[c][i * 3 + 0] = pi0 * r00 + pi1 * r10 + pi2 * r20;
        R[c][i * 3 + 1] = pi0 * r01 + pi1 * r11 + pi2 * r21;
        R[c][i * 3 + 2] = pi0 * r02 + pi1 * r12 + pi2 * r22;
        T[c][i]         = pi0 * ltx + pi1 * lty + pi2 * ltz + T[p][i];
      }
    }
    op[c * 3 + 0] = T[c][0];
    op[c * 3 + 1] = T[c][1];
    op[c * 3 + 2] = T[c][2];
  }
}

// ---- Hot kernel: B % TPB == 0, no tail code --------------------------------
__global__ __launch_bounds__(TPB) void Skeleton_fk_full(
    const float* __restrict__ angles, const float* __restrict__ xyz,
    float* __restrict__ out) {
  __shared__ float ang[TPB * STRIDE];   // 151552 B: padded per-element tile
  __shared__ float xyzS[NJ * 3];

  const int tid = threadIdx.x;
  const int e0  = blockIdx.x * TPB;
  const float* gbase = angles + (size_t)e0 * ELEM_F;

  // Issue all async copies first (one /36, then incremental address updates).
  // chunk q = c*TPB + tid; delta per iter: 256 = 7*36 + 4.
  {
    const int eI = tid / CHUNKS;                     // element within tile
    int rr = tid - eI * CHUNKS;                      // chunk within element
    unsigned loff = (unsigned)(eI * (STRIDE * 4) + rr * 16);   // LDS byte offset
    const float* src = gbase + tid * 4;
#pragma unroll
    for (int c = 0; c < CHUNKS; ++c) {
      async_copy_b128((float*)((char*)ang + loff), src);
      src  += TPB * 4;                               // +4096 B global
      rr   += 4;
      loff += 7u * (STRIDE * 4) + 4 * 16;            // +4208 B
      if (rr >= CHUNKS) { rr -= CHUNKS; loff += (STRIDE - ELEM_F) * 4; }  // +16 B
    }
  }

  // Stage reference xyz while the DMA traffic is in flight.
  if (tid < NJ * 3) xyzS[tid] = xyz[tid];

  async_wait0();
  __syncthreads();

  const int e = e0 + tid;
  fk_elem(&ang[tid * STRIDE], xyzS, out + (size_t)e * OUT_F);
}

// ---- Generic kernel: any B, direct global reads ----------------------------
__global__ __launch_bounds__(TPB) void Skeleton_fk_any(
    const float* __restrict__ angles, const float* __restrict__ xyz,
    float* __restrict__ out, int B) {
  __shared__ float xyzS[NJ * 3];
  const int tid = threadIdx.x;
  if (tid < NJ * 3) xyzS[tid] = xyz[tid];
  __syncthreads();
  const int e = blockIdx.x * TPB + tid;
  if (e < B) fk_elem(angles + (size_t)e * ELEM_F, xyzS, out + (size_t)e * OUT_F);
}

extern "C" void kernel_launch(void* const* d_in, const int* in_sizes, int n_in,
                              void* d_out, int out_size, void* d_ws, size_t ws_size,
                              hipStream_t stream) {
  const float* angles = (const float*)d_in[0];
  const float* xyz    = (const float*)d_in[1];
  float* out          = (float*)d_out;
  const int B = in_sizes[0] / ELEM_F;        // 131072
  if (B % TPB == 0) {
    Skeleton_fk_full<<<dim3(B / TPB), dim3(TPB), 0, stream>>>(angles, xyz, out);
  } else {
    Skeleton_fk_any<<<dim3((B + TPB - 1) / TPB), dim3(TPB), 0, stream>>>(angles, xyz, out, B);
  }
  (void)n_in; (void)out_size; (void)d_ws; (void)ws_size;
}